// YOLOv1Loss_8641474199685
// MI455X (gfx1250) — compile-verified
//
#include <hip/hip_runtime.h>
#include <hip/hip_bf16.h>

// YOLOv1 loss for MI455X (gfx1250, wave32).
// Bandwidth-bound: 193 MB read @ 23.3 TB/s => ~8.3us floor. Strategy:
//  - LDS-staged tiles (256 cells/block) so every global load is a fully
//    coalesced 128B cacheline per wave32.
//  - stride-30 LDS reads are bank-conflict-free under wave32.
//  - wave reduction via V_WMMA_F32_16X16X4_F32 (B = ones): D[m][n] =
//    x_m + x_{m+16}; lane-local sum of 8 D regs + shfl(16) = 32-lane sum.
//  - deterministic two-pass block reduction (no float atomics).

typedef __attribute__((ext_vector_type(2))) float v2f;
typedef __attribute__((ext_vector_type(8))) float v8f;

#define TILE 256
#define LAST 30
#define NCELLS (16384 * 7 * 7)   // 802816
#define NBLK (NCELLS / TILE)     // 3136 exactly

__device__ __forceinline__ float wave_reduce_wmma(float x) {
  // A (16x4 f32): vgpr0 = x, vgpr1 = 0.
  // Lanes 0-15 supply A[m][0], lanes 16-31 supply A[m][2]  (m = lane%16).
  v2f a; a.x = x;    a.y = 0.0f;
  v2f b; b.x = 1.0f; b.y = 1.0f;          // B = all ones
  v8f c = {};
  v8f d = __builtin_amdgcn_wmma_f32_16x16x4_f32(
      /*neg_a=*/false, a, /*neg_b=*/false, b,
      /*c_mod=*/(short)0, c, /*reuse_a=*/false, /*reuse_b=*/false);
  // Lane L<16 holds D rows 0..7 of column L; lane L>=16 holds rows 8..15.
  float s = d[0] + d[1] + d[2] + d[3] + d[4] + d[5] + d[6] + d[7];
  // lanes 0..15: s = sum_{m=0..7}(x_m + x_{m+16}); add lanes-16..31 half.
  s += __shfl(s, 16, 32);
  return s;  // correct full 32-lane sum in lanes 0..15
}

__device__ __forceinline__ float iou_center(float cx1, float cy1, float w1, float h1,
                                            float cx2, float cy2, float w2, float h2) {
  float x1min = cx1 - 0.5f * w1, x1max = cx1 + 0.5f * w1;
  float y1min = cy1 - 0.5f * h1, y1max = cy1 + 0.5f * h1;
  float x2min = cx2 - 0.5f * w2, x2max = cx2 + 0.5f * w2;
  float y2min = cy2 - 0.5f * h2, y2max = cy2 + 0.5f * h2;
  float iw = fmaxf(fminf(x1max, x2max) - fmaxf(x1min, x2min), 0.0f);
  float ih = fmaxf(fminf(y1max, y2max) - fmaxf(y1min, y2min), 0.0f);
  float inter = iw * ih;
  return inter / (w1 * h1 + w2 * h2 - inter + 1e-10f);
}

__global__ __launch_bounds__(TILE) void yolo_partial_kernel(
    const float* __restrict__ preds, const float* __restrict__ labels,
    float* __restrict__ part) {
  __shared__ float lp[TILE * LAST];
  __shared__ float ll[TILE * LAST];
  __shared__ float wsum[TILE / 32];

  const int t = threadIdx.x;
  const int base = blockIdx.x * (TILE * LAST);  // < 24.1M, fits int

  // Coalesced staging: each global_load_b32 covers exactly one 128B line/wave.
#pragma unroll
  for (int i = 0; i < LAST; ++i) {
    const int idx = t + i * TILE;
    lp[idx] = preds[base + idx];
    ll[idx] = labels[base + idx];
  }
  __syncthreads();

  const float* p = &lp[t * LAST];
  const float* l = &ll[t * LAST];

  const float iou1 = iou_center(p[0], p[1], p[2], p[3], l[0], l[1], l[2], l[3]);
  const float iou2 = iou_center(p[5], p[6], p[7], p[8], l[0], l[1], l[2], l[3]);
  const bool use1 = iou1 > iou2;

  float dx, dy, dw, dh, objd, noobj;
  if (use1) {
    dx = l[0] - p[0];  dy = l[1] - p[1];
    dw = sqrtf(l[2]) - sqrtf(p[2]);
    dh = sqrtf(l[3]) - sqrtf(p[3]);
    objd = iou1 - p[4];
    noobj = p[9];
  } else {
    dx = l[5] - p[5];  dy = l[6] - p[6];
    dw = sqrtf(l[7]) - sqrtf(p[7]);
    dh = sqrtf(l[8]) - sqrtf(p[8]);
    objd = iou2 - p[9];
    noobj = p[4];
  }
  const float coord_xy = 5.0f * (dx * dx + dy * dy);
  const float coord_wh = dw * dw + dh * dh;
  const float obj_loss = objd * objd;
  const float noobj_resp = 0.5f * noobj * noobj;

  float cls = 0.0f;
#pragma unroll
  for (int cidx = 10; cidx < 30; ++cidx) {
    const float dd = l[cidx] - p[cidx];
    cls += dd * dd;
  }

  const float noobj_cell = 0.5f * (p[4] * p[4] + p[9] * p[9]);
  const float cell = (l[4] == 1.0f)
      ? (coord_xy + coord_wh + obj_loss + noobj_resp + cls)
      : noobj_cell;

  // Per-wave reduction via WMMA (EXEC all-ones here: no divergence).
  const float s = wave_reduce_wmma(cell);
  if ((t & 31) == 0) wsum[t >> 5] = s;
  __syncthreads();
  if (t == 0) {
    float tot = 0.0f;
#pragma unroll
    for (int w = 0; w < TILE / 32; ++w) tot += wsum[w];
    part[blockIdx.x] = tot;
  }
}

__global__ __launch_bounds__(TILE) void yolo_finalize_kernel(
    const float* __restrict__ part, float* __restrict__ out) {
  __shared__ float wsum[TILE / 32];
  const int t = threadIdx.x;
  float acc = 0.0f;
  for (int i = t; i < NBLK; i += TILE) acc += part[i];  // fixed order: deterministic
  const float s = wave_reduce_wmma(acc);
  if ((t & 31) == 0) wsum[t >> 5] = s;
  __syncthreads();
  if (t == 0) {
    float tot = 0.0f;
#pragma unroll
    for (int w = 0; w < TILE / 32; ++w) tot += wsum[w];
    out[0] = tot * (1.0f / 16384.0f);
  }
}

extern "C" void kernel_launch(void* const* d_in, const int* in_sizes, int n_in,
                              void* d_out, int out_size, void* d_ws, size_t ws_size,
                              hipStream_t stream) {
  const float* preds  = (const float*)d_in[0];
  const float* labels = (const float*)d_in[1];
  float* part = (float*)d_ws;      // NBLK floats = 12.5 KB of scratch
  float* out  = (float*)d_out;

  yolo_partial_kernel<<<NBLK, TILE, 0, stream>>>(preds, labels, part);
  yolo_finalize_kernel<<<1, TILE, 0, stream>>>(part, out);
}